// GaussianVAE_41747082117131
// MI455X (gfx1250) — compile-verified
//
#include <hip/hip_runtime.h>
#include <hip/hip_bf16.h>

// Problem constants (match reference): B=8, N=8192, D=3, G=512
#define BB 8
#define NN 8192
#define GG 512

typedef __attribute__((ext_vector_type(2))) float v2f;
typedef __attribute__((ext_vector_type(4))) float v4f;
typedef __attribute__((ext_vector_type(8))) float v8f;

// Block layout: 256 threads = 8 waves (wave32). Each wave owns 16 predicted
// points (n columns); a block covers 128 n points of one batch.
// LDS: 8192 real points as float4 (x, y, z, |r|^2) = 128KB, + 128 winner ints.
#define THREADS 256
#define NPB 128              // n points per block
#define NBLK (NN / NPB)      // 64 blocks per batch
#define SMEM_BYTES (NN * 16 + NPB * 4)

__global__ void nearest_expr_kernel(const float* __restrict__ pred,
                                    const float* __restrict__ realp,
                                    const float* __restrict__ expr,
                                    float* __restrict__ out) {
    extern __shared__ char smem_raw[];
    v4f* lds4 = (v4f*)smem_raw;                    // [NN] real points (x,y,z,sq)
    int* idxs = (int*)(smem_raw + NN * 16);        // [NPB] winner indices

    const int tid  = threadIdx.x;
    const int lane = tid & 31;
    const int w    = tid >> 5;
    const int l16  = lane & 15;
    const int hi   = (lane >> 4);                  // 0 for lanes 0-15, 1 for 16-31
    const int hi8  = hi << 3;

    const int b      = blockIdx.x / NBLK;
    const int nblock = blockIdx.x % NBLK;
    const int nbase  = nblock * NPB;

    // ---- Stage real positions (+ squared norm) into LDS ----
    {
        const float* rbase = realp + (size_t)b * NN * 3;
        for (int p = tid; p < NN; p += THREADS) {
            const float* rp = rbase + (size_t)p * 3;
            float x = rp[0], y = rp[1], z = rp[2];
            v4f v;
            v.x = x; v.y = y; v.z = z; v.w = x * x + y * y + z * z;
            lds4[p] = v;
        }
    }

    // ---- B fragment: constant per wave. Column n = nbase + w*16 + l16.
    // B is K=4 x N=16; per-lane 2 VGPRs (VGPR0: K=0/K=2, VGPR1: K=1/K=3 for
    // lane halves). Column n = (-2px, -2py, -2pz, 1) so
    // D[m][n] = |r|^2 - 2 p.r  (argmin-equivalent score; p^2 drops out).
    const int ncol = nbase + w * 16 + l16;
    const float* pp = pred + ((size_t)b * NN + (size_t)ncol) * 3;
    const float n2px = -2.0f * pp[0];
    const float n2py = -2.0f * pp[1];
    const float n2pz = -2.0f * pp[2];
    v2f bfrag;
    bfrag.x = hi ? n2pz : n2px;
    bfrag.y = hi ? 1.0f : n2py;

    __syncthreads();

    // ---- Main loop: value-only tile-min + winning-tile tracking ----
    float best  = __builtin_inff();
    int   btile = 0;
    const char* smem_bytes = smem_raw;
    unsigned aoff = (unsigned)(l16 * 16 + (hi ? 8 : 0));  // running LDS byte offset

    for (int mbase = 0; mbase < NN; mbase += 16) {
        // A fragment (16x4 f32): lanes 0-15 hold M=lane K={0,1} (x,y);
        // lanes 16-31 hold M=lane-16 K={2,3} (z,|r|^2).
        v2f a = *(const v2f*)(smem_bytes + aoff);
        aoff += 256;

        v8f c = {};
        v8f d = __builtin_amdgcn_wmma_f32_16x16x4_f32(
            false, a, false, bfrag, (short)0, c, false, false);

        // min-tree over the 8 D slots (compiler fuses to v_min3_num_f32)
        float m0 = __builtin_fminf(d[0], d[1]);
        float m1 = __builtin_fminf(d[2], d[3]);
        float m2 = __builtin_fminf(d[4], d[5]);
        float m3 = __builtin_fminf(d[6], d[7]);
        float tmin = __builtin_fminf(__builtin_fminf(m0, m1),
                                     __builtin_fminf(m2, m3));
        bool t = tmin < best;               // strict < keeps earliest tile
        best  = t ? tmin : best;
        btile = t ? mbase : btile;
    }

    // ---- Recover within-tile index: scalar-recompute the one winning tile.
    // This lane's candidate set was m = btile + hi8 + r, r in 0..7.
    float bv = __builtin_inff();
    int   bidx = 0;
#pragma unroll
    for (int r = 0; r < 8; ++r) {
        v4f rp = lds4[btile + hi8 + r];
        float s = __builtin_fmaf(rp.x, n2px,
                  __builtin_fmaf(rp.y, n2py,
                  __builtin_fmaf(rp.z, n2pz, rp.w)));
        bool t = s < bv;                    // keep-first on ties
        bidx = t ? (btile + hi8 + r) : bidx;
        bv   = t ? s : bv;
    }

    // ---- Merge the two lane halves (same n column, disjoint m halves) ----
    {
        float ov = __shfl_xor(bv, 16, 32);
        int   oi = __shfl_xor(bidx, 16, 32);
        bool take = (ov < bv) || (ov == bv && oi < bidx);
        bidx = take ? oi : bidx;
        if (lane < 16) idxs[w * 16 + l16] = bidx;
    }
    __syncthreads();

    // ---- Gather: out[b, nbase+row, :] = expr[b, idxs[row], :]  (2KB/row) ----
    {
        const float* ebase = expr + (size_t)b * NN * GG;
        float* obase = out + ((size_t)b * NN + (size_t)nbase) * GG;
        // 128 rows x 128 float4 chunks = 16384 chunks, 256 threads -> 64 iters
        for (int it = tid; it < NPB * (GG / 4); it += THREADS) {
            int row = it >> 7;
            int c4  = it & 127;
            int src = idxs[row];
            v4f val = *(const v4f*)(ebase + (size_t)src * GG + (size_t)c4 * 4);
            *(v4f*)(obase + (size_t)row * GG + (size_t)c4 * 4) = val;
        }
    }
}

extern "C" void kernel_launch(void* const* d_in, const int* in_sizes, int n_in,
                              void* d_out, int out_size, void* d_ws, size_t ws_size,
                              hipStream_t stream) {
    const float* pred  = (const float*)d_in[0];  // (B, N, 3) f32
    const float* realp = (const float*)d_in[1];  // (B, N, 3) f32
    const float* expr  = (const float*)d_in[2];  // (B, N, G) f32
    float* out = (float*)d_out;                  // (B, N, G) f32

    dim3 grid(BB * NBLK);   // 512 blocks
    dim3 block(THREADS);    // 256 threads = 8 waves
    nearest_expr_kernel<<<grid, block, SMEM_BYTES, stream>>>(pred, realp, expr, out);
}